// KDNet_Batch_71150428226023
// MI455X (gfx1250) — compile-verified
//
#include <hip/hip_runtime.h>
#include <hip/hip_bf16.h>

typedef __attribute__((ext_vector_type(2))) float v2f;
typedef __attribute__((ext_vector_type(8))) float v8f;

#define BSZ 128

// ---------------------------------------------------------------------------
// Fast GEMM (levels 1-10), fully constant-shaped: DIM = 1<<LDIM, CIN%4==0,
// MM%16==0. One wave32 computes a 16(M) x 64(N) tile = 4 WMMA accumulators,
// reusing the A (weight) fragment across the 4 N-subtiles. Constant shapes ->
// strides fold into load immediates / fixed pointer bumps; constant trip
// count -> no remainder path.
// A lane layout: lanes 0-15 hold rows m0..m0+15 with K=(k0,k0+1) packed v2f;
// lanes 16-31 the same rows with K=(k0+2,k0+3).
// D layout: vgpr r -> (M = m0 + r + 8*(lane>=16), N = ntile + (lane&15)).
// ---------------------------------------------------------------------------
template <int LDIM, int CIN, int MM>
__global__ __launch_bounds__(32) void kd_gemm_wmma_fast(
    const float* __restrict__ X, const float* __restrict__ W,
    const float* __restrict__ bias, float* __restrict__ Y)
{
  constexpr int DIM = 1 << LDIM;
  const int lane = threadIdx.x & 31;
  const int nl   = lane & 15;
  const int kh   = lane >> 4;
  const int m0   = blockIdx.y * 16;
  const int n0   = blockIdx.x * 64;

  int bcol[4], lcol[4];
  const float* xk[4];
#pragma unroll
  for (int j = 0; j < 4; ++j) {
    int n = n0 + j * 16 + nl;
    int b = n >> LDIM;
    int l = n & (DIM - 1);
    bcol[j] = b;
    lcol[j] = l;
    // fold the lane's K-pair offset (2*kh rows of X) into the base pointer
    xk[j] = X + (size_t)b * (CIN * DIM) + l + (size_t)(2 * kh) * DIM;
  }
  const float* wr = W + (size_t)(m0 + nl) * CIN + 2 * kh;  // 8B aligned

  v8f acc0 = {}, acc1 = {}, acc2 = {}, acc3 = {};
#pragma unroll 2
  for (int k0 = 0; k0 < CIN; k0 += 4) {
    v2f Av = *(const v2f*)(wr + k0);                       // packed K pair
    v2f B0 = { xk[0][(size_t)k0 * DIM], xk[0][(size_t)k0 * DIM + DIM] };
    v2f B1 = { xk[1][(size_t)k0 * DIM], xk[1][(size_t)k0 * DIM + DIM] };
    v2f B2 = { xk[2][(size_t)k0 * DIM], xk[2][(size_t)k0 * DIM + DIM] };
    v2f B3 = { xk[3][(size_t)k0 * DIM], xk[3][(size_t)k0 * DIM + DIM] };
    acc0 = __builtin_amdgcn_wmma_f32_16x16x4_f32(false, Av, false, B0, (short)0, acc0, false, false);
    acc1 = __builtin_amdgcn_wmma_f32_16x16x4_f32(false, Av, false, B1, (short)0, acc1, false, false);
    acc2 = __builtin_amdgcn_wmma_f32_16x16x4_f32(false, Av, false, B2, (short)0, acc2, false, false);
    acc3 = __builtin_amdgcn_wmma_f32_16x16x4_f32(false, Av, false, B3, (short)0, acc3, false, false);
  }

  const int mD = m0 + 8 * kh;
  float bi[8];
#pragma unroll
  for (int r = 0; r < 8; ++r) bi[r] = bias[mD + r];

  {
    float* yb = Y + (((size_t)bcol[0] * MM + mD) << LDIM) + lcol[0];
#pragma unroll
    for (int r = 0; r < 8; ++r) yb[(size_t)r << LDIM] = acc0[r] + bi[r];
  }
  {
    float* yb = Y + (((size_t)bcol[1] * MM + mD) << LDIM) + lcol[1];
#pragma unroll
    for (int r = 0; r < 8; ++r) yb[(size_t)r << LDIM] = acc1[r] + bi[r];
  }
  {
    float* yb = Y + (((size_t)bcol[2] * MM + mD) << LDIM) + lcol[2];
#pragma unroll
    for (int r = 0; r < 8; ++r) yb[(size_t)r << LDIM] = acc2[r] + bi[r];
  }
  {
    float* yb = Y + (((size_t)bcol[3] * MM + mD) << LDIM) + lcol[3];
#pragma unroll
    for (int r = 0; r < 8; ++r) yb[(size_t)r << LDIM] = acc3[r] + bi[r];
  }
}

// ---------------------------------------------------------------------------
// Level-0 GEMM, specialized: Cin=3, M=24, dim=2048. Single K-step -> one WMMA.
// All guards resolved at compile time into clamped addresses + value selects
// (unconditional in-bounds loads + v_cndmask), no exec divergence before the
// WMMA. Store: mD in {0,8,16,24}; rows valid iff mD < 24, and then all 8 are.
// ---------------------------------------------------------------------------
__global__ __launch_bounds__(32) void kd_gemm_wmma_l0(
    const float* __restrict__ X, const float* __restrict__ W,
    const float* __restrict__ bias, float* __restrict__ Y)
{
  constexpr int DIM = 2048, M = 24;
  const int lane = threadIdx.x & 31;
  const int nl   = lane & 15;
  const int kh   = lane >> 4;
  const int n0   = blockIdx.x * 16;
  const int m0   = blockIdx.y * 16;

  const int n = n0 + nl;
  const int b = n >> 11;
  const int l = n & (DIM - 1);
  const int mA = m0 + nl;
  const bool mok = (mA < M);

  const float* wr = W + (size_t)(mok ? mA : 0) * 3;  // clamped, in-bounds
  const float* xb = X + (size_t)b * 3 * DIM + l;

  // K indices: lanes 0-15 -> (0,1); lanes 16-31 -> (2,3); K=3 is zero-pad.
  float wa0 = wr[2 * kh];                 // K = 0 or 2, always < 3
  float wa1 = wr[1];                      // reused as K=1 value for kh==0
  float xa0 = xb[(size_t)(2 * kh) * DIM]; // K = 0 or 2
  float xa1 = xb[(size_t)DIM];            // K = 1 value (in-bounds)

  float a0 = mok ? wa0 : 0.0f;
  float a1 = (mok && kh == 0) ? wa1 : 0.0f;
  float b0 = xa0;
  float b1 = (kh == 0) ? xa1 : 0.0f;

  v2f Av = {a0, a1};
  v2f Bv = {b0, b1};
  v8f acc = {};
  acc = __builtin_amdgcn_wmma_f32_16x16x4_f32(
      false, Av, false, Bv, (short)0, acc, false, false);

  const int mD = m0 + 8 * kh;
  if (mD < M) {                            // mD in {0,8,16} -> all 8 rows valid
    float* yb = Y + (((size_t)b * M + mD) << 11) + l;
#pragma unroll
    for (int r = 0; r < 8; ++r) yb[(size_t)r << 11] = acc[r] + bias[mD + r];
  }
}

// ---------------------------------------------------------------------------
// Per-channel batch statistics (training-mode BN, biased variance).
// ---------------------------------------------------------------------------
__global__ __launch_bounds__(256) void kd_bn_stats(
    const float* __restrict__ Y, float* __restrict__ mean,
    float* __restrict__ rstd, int M, int ldim)
{
  __shared__ float ssum[256];
  __shared__ float ssq[256];
  const int o = blockIdx.x;
  const int dmask = (1 << ldim) - 1;
  const int N = BSZ << ldim;
  float s = 0.0f, q = 0.0f;
  for (int i = threadIdx.x; i < N; i += 256) {
    int bb = i >> ldim;
    int ll = i & dmask;
    float v = Y[(((size_t)bb * M + o) << ldim) + ll];
    s += v;
    q += v * v;
  }
  ssum[threadIdx.x] = s;
  ssq[threadIdx.x]  = q;
  __syncthreads();
  for (int st = 128; st > 0; st >>= 1) {
    if (threadIdx.x < st) {
      ssum[threadIdx.x] += ssum[threadIdx.x + st];
      ssq[threadIdx.x]  += ssq[threadIdx.x + st];
    }
    __syncthreads();
  }
  if (threadIdx.x == 0) {
    float inv = 1.0f / (float)N;
    float mu  = ssum[0] * inv;
    float var = ssq[0] * inv - mu * mu;
    mean[o] = mu;
    rstd[o] = rsqrtf(var + 1e-5f);
  }
}

// ---------------------------------------------------------------------------
// Fused BN affine + ReLU + bug-faithful gather + pair/reshape.
//   f = c2 % F; d = 2p + (c2 >= F); j = 3d + sel[b,d];
//   o = 3f + j/dim; l = j%dim.   (F and dim are powers of two)
// ---------------------------------------------------------------------------
__global__ __launch_bounds__(256) void kd_bn_gather(
    const float* __restrict__ Y, const int* __restrict__ sel,
    const float* __restrict__ mean, const float* __restrict__ rstd,
    const float* __restrict__ g, const float* __restrict__ beta,
    float* __restrict__ Xn, int lF, int ldim)
{
  const int F    = 1 << lF;
  const int dim  = 1 << ldim;
  const int lhalf = ldim - 1;
  const int lC2   = lF + 1;
  const long total = (long)BSZ << (lC2 + lhalf);
  long tid = (long)blockIdx.x * 256 + threadIdx.x;
  if (tid >= total) return;

  int p  = (int)(tid & ((1 << lhalf) - 1));
  int c2 = (int)((tid >> lhalf) & ((2 << lF) - 1));
  int b  = (int)(tid >> (lhalf + lC2));

  int f = c2 & (F - 1);
  int d = 2 * p + (c2 >> lF);
  int j = 3 * d + sel[((size_t)b << ldim) + d];
  int jd = j >> ldim;
  int o = 3 * f + jd;
  int l = j & (dim - 1);
  int M = 3 * F;

  float v = (Y[(((size_t)b * M + o) << ldim) + l] - mean[o]) * rstd[o] * g[o] + beta[o];
  Xn[tid] = v > 0.0f ? v : 0.0f;
}

// ---------------------------------------------------------------------------
// Final FC (2048 -> 16) + log_softmax, one block per batch row.
// ---------------------------------------------------------------------------
__global__ __launch_bounds__(256) void kd_fc_logsoftmax(
    const float* __restrict__ H, const float* __restrict__ Wf,
    const float* __restrict__ bf, float* __restrict__ out)
{
  __shared__ float part[256 * 16];
  __shared__ float logits[16];
  __shared__ float lse;
  const int b = blockIdx.x;
  const int t = threadIdx.x;

  float acc[16];
#pragma unroll
  for (int k = 0; k < 16; ++k) acc[k] = 0.0f;
  for (int i = t; i < 2048; i += 256) {
    float h = H[(size_t)b * 2048 + i];
#pragma unroll
    for (int k = 0; k < 16; ++k) acc[k] += h * Wf[(size_t)k * 2048 + i];
  }
#pragma unroll
  for (int k = 0; k < 16; ++k) part[t * 16 + k] = acc[k];
  __syncthreads();

  if (t < 16) {
    float s = 0.0f;
    for (int i = 0; i < 256; ++i) s += part[i * 16 + t];
    logits[t] = s + bf[t];
  }
  __syncthreads();
  if (t == 0) {
    float m = logits[0];
    for (int k = 1; k < 16; ++k) m = fmaxf(m, logits[k]);
    float s = 0.0f;
    for (int k = 0; k < 16; ++k) s += expf(logits[k] - m);
    lse = m + logf(s);
  }
  __syncthreads();
  if (t < 16) out[b * 16 + t] = logits[t] - lse;
}

// ---------------------------------------------------------------------------
// Host orchestration.
// ---------------------------------------------------------------------------
extern "C" void kernel_launch(void* const* d_in, const int* in_sizes, int n_in,
                              void* d_out, int out_size, void* d_ws, size_t ws_size,
                              hipStream_t stream) {
  static const int LDIM[11] = {11, 10, 9, 8, 7, 6, 5, 4, 3, 2, 1};   // log2(dim)
  static const int LF[11]   = {3, 5, 6, 6, 6, 7, 8, 9, 9, 9, 10};    // log2(F)

  const float* x = (const float*)d_in[0];
  const int*   sels[11];
  const float* cw[11];
  const float* cb[11];
  const float* bg[11];
  const float* bb[11];
  for (int i = 0; i < 11; ++i) {
    sels[i] = (const int*)d_in[1 + i];
    cw[i]   = (const float*)d_in[12 + i];
    cb[i]   = (const float*)d_in[23 + i];
    bg[i]   = (const float*)d_in[34 + i];
    bb[i]   = (const float*)d_in[45 + i];
  }
  const float* fcw = (const float*)d_in[56];
  const float* fcb = (const float*)d_in[57];

  // Workspace layout (bytes):
  //   Y  : [0, 48 MiB) | X0 : [48,64) MiB | X1 : [64,80) MiB | mean/rstd after
  char* ws = (char*)d_ws;
  float* Ybuf  = (float*)(ws);
  float* X0    = (float*)(ws + (size_t)48 * 1024 * 1024);
  float* X1    = (float*)(ws + (size_t)64 * 1024 * 1024);
  float* meanb = (float*)(ws + (size_t)80 * 1024 * 1024);
  float* rstdb = (float*)(ws + (size_t)80 * 1024 * 1024 + 16384);

#define LAUNCH_FAST(LD, CI, MM3)                                          \
  kd_gemm_wmma_fast<LD, CI, MM3>                                          \
      <<<dim3((BSZ << (LD)) / 64, (MM3) / 16), 32, 0, stream>>>(          \
          Xcur, cw[lvl], cb[lvl], Ybuf)

  const float* Xcur = x;
  for (int lvl = 0; lvl < 11; ++lvl) {
    const int ldim = LDIM[lvl];
    const int lF   = LF[lvl];
    const int F    = 1 << lF;
    const int M    = 3 * F;
    float* Xn = (lvl & 1) ? X1 : X0;

    switch (lvl) {
      case 0:
        kd_gemm_wmma_l0<<<dim3((BSZ << 11) / 16, 2), 32, 0, stream>>>(
            Xcur, cw[0], cb[0], Ybuf);
        break;
      case 1:  LAUNCH_FAST(10, 16,   96);   break;
      case 2:  LAUNCH_FAST(9,  64,   192);  break;
      case 3:  LAUNCH_FAST(8,  128,  192);  break;
      case 4:  LAUNCH_FAST(7,  128,  192);  break;
      case 5:  LAUNCH_FAST(6,  128,  384);  break;
      case 6:  LAUNCH_FAST(5,  256,  768);  break;
      case 7:  LAUNCH_FAST(4,  512,  1536); break;
      case 8:  LAUNCH_FAST(3,  1024, 1536); break;
      case 9:  LAUNCH_FAST(2,  1024, 1536); break;
      case 10: LAUNCH_FAST(1,  1024, 3072); break;
    }

    kd_bn_stats<<<M, 256, 0, stream>>>(Ybuf, meanb, rstdb, M, ldim);

    long total = (long)BSZ << (lF + ldim);   // B * 2F * (dim/2)
    kd_bn_gather<<<(unsigned)((total + 255) / 256), 256, 0, stream>>>(
        Ybuf, sels[lvl], meanb, rstdb, bg[lvl], bb[lvl], Xn, lF, ldim);

    Xcur = Xn;
  }

  kd_fc_logsoftmax<<<BSZ, 256, 0, stream>>>(Xcur, fcw, fcb, (float*)d_out);
#undef LAUNCH_FAST
}